// ConvLRU_2697239462379
// MI455X (gfx1250) — compile-verified
//
#include <hip/hip_runtime.h>
#include <hip/hip_bf16.h>

// ---------------------------------------------------------------------------
// ConvLRU on MI455X (gfx1250, wave32, WMMA + TDM).
// Pipeline (einsum commuted with FFT to halve GEMM count):
//   1. params:  lam = exp(-exp(nu_log))*e^{i exp(theta_log)}, gamma = exp(gamma_log)
//   2. mixB:    Dre = Wb_re*x, Dim = Wb_im*x        (f32 WMMA GEMM, TDM-staged LDS)
//   3. fft2:    (Dre+iDim) -> freq domain, epilogue (+bb)*gamma   (in place)
//   4. scan:    y_l = lam*y_{l-1} + h_l over L      (in place; emits last_hidden)
//   5. ifft2:   back to spatial domain               (in place)
//   6. mixC:    E = Wc_re*Sre - Wc_im*Sim + bc_re   (f32 WMMA GEMM, TDM-staged LDS)
//   7. LN+res:  layernorm over (C,H,W) + x -> out
// ---------------------------------------------------------------------------

typedef __attribute__((ext_vector_type(2))) float v2f;
typedef __attribute__((ext_vector_type(8))) float v8f;
typedef __attribute__((ext_vector_type(4))) unsigned int u32x4;
typedef __attribute__((ext_vector_type(8))) unsigned int u32x8;

namespace {
constexpr int kB = 2, kL = 32, kC = 128, kH = 32, kW = 32;
constexpr int kHW = kH * kW;                  // 1024
constexpr int kN1 = kB * kL * kC * kHW;       // 8,388,608 floats per plane
constexpr int kCH = kC * kH;                  // 4096
}

// ------------------------------------------------------ TDM 2D tile load ----
// Issue a Tensor Data Mover load of a (tile_d1 x tile_d0) f32 tile from a
// row-major (d1 x d0_stride) f32 tensor into LDS, padding the LDS destination
// by 1 DWORD after each row (pad_interval code: 2<<code DWORDs per row).
// Descriptor layout per CDNA5 ISA.
__device__ __forceinline__ void tdm_load_tile_f32(
        unsigned lds_byte_addr, unsigned long long gaddr_bytes,
        unsigned tensor_d0, unsigned tensor_d1,
        unsigned long long d0_stride_elems,
        unsigned tile_d0, unsigned tile_d1,
        unsigned pad_interval_code) {
    u32x4 g0;
    g0.x = 1u;                                             // count=1, user D#
    g0.y = lds_byte_addr;                                  // [63:32] lds_addr
    g0.z = (unsigned)(gaddr_bytes & 0xffffffffu);          // [95:64] ga lo
    g0.w = (unsigned)((gaddr_bytes >> 32) & 0x01ffffffu)   // [120:96] ga hi
         | 0x80000000u;                                    // [127:126] type=2
    u32x8 g1;
    g1.s0 = (2u << 16)                                     // data_size = 4B
          | (1u << 20)                                     // pad_enable
          | (pad_interval_code << 22)                      // pad interval
          | (0u << 25);                                    // pad amount = 1 DW
    g1.s1 = (tensor_d0 & 0xffffu) << 16;                   // dim0[15:0]
    g1.s2 = (tensor_d0 >> 16) | ((tensor_d1 & 0xffffu) << 16);
    g1.s3 = (tensor_d1 >> 16) | (tile_d0 << 16);           // tile_dim0
    g1.s4 = (tile_d1 & 0xffffu);                           // tile_dim1, dim2=0
    g1.s5 = (unsigned)(d0_stride_elems & 0xffffffffu);     // dim0_stride lo
    g1.s6 = (unsigned)((d0_stride_elems >> 32) & 0xffffu); // dim0_stride hi
    g1.s7 = 0u;                                            // dim1_stride unused
    asm volatile("tensor_load_to_lds %0, %1" :: "s"(g0), "s"(g1) : "memory");
}

// ---------------------------------------------------------------- params ----
__global__ void convlru_params(const float* __restrict__ plog,
                               float* __restrict__ lamRe,
                               float* __restrict__ lamIm,
                               float* __restrict__ gam) {
    int i = blockIdx.x * blockDim.x + threadIdx.x;
    if (i >= kCH) return;
    float nu = __expf(plog[i]);
    float th = __expf(plog[kCH + i]);
    float g  = __expf(plog[2 * kCH + i]);
    float r  = __expf(-nu);
    float s, c;
    __sincosf(th, &s, &c);
    lamRe[i] = r * c;
    lamIm[i] = r * s;
    gam[i]   = g;
}

// ------------------------------------------------------------- mixB GEMM ----
// D{re,im}[bl, o, hw] = sum_c W{re,im}[o,c] * x[bl, c, hw]
// Block = 256 threads (8 waves); tile = 128 rows x 64 cols; wave w owns rows
// [16w,16w+16); 4 N-subtiles of 16; K-loop in steps of 4 via wmma f32 16x16x4.
// X tile staged global->LDS by the Tensor Data Mover (pad -> stride 65).
#define LDSPB 65
__global__ __launch_bounds__(256) void convlru_mixb(
        const float* __restrict__ x,
        const float* __restrict__ Wre, const float* __restrict__ Wim,
        float* __restrict__ Dre, float* __restrict__ Dim) {
    __shared__ float Xs[128 * LDSPB];
    const int bl   = blockIdx.y;
    const int col0 = blockIdx.x * 64;

    if (threadIdx.x < 32) {       // wave 0 issues the DMA and awaits TENSORcnt
        unsigned long long ga =
            (unsigned long long)(const void*)(x + (size_t)bl * kC * kHW + col0);
        tdm_load_tile_f32((unsigned)(unsigned long long)(const void*)Xs, ga,
                          kHW, kC, kHW, 64u, 128u, /*pad 64DW*/ 5u);
        __builtin_amdgcn_s_wait_tensorcnt(0);
    }
    __syncthreads();

    const int wave = threadIdx.x >> 5;
    const int lane = threadIdx.x & 31;
    const int rowBase = wave * 16;
    const int m = lane & 15;         // A row-in-tile / B col-in-tile
    const int half = lane >> 4;      // K-half selector per ISA layout

    v8f accRe[4] = {}, accIm[4] = {};

    for (int k = 0; k < 128; k += 4) {
        v2f aRe, aIm;
        int arow = (rowBase + m) * kC + k + 2 * half;
        aRe.x = Wre[arow]; aRe.y = Wre[arow + 1];
        aIm.x = Wim[arow]; aIm.y = Wim[arow + 1];
#pragma unroll
        for (int nt = 0; nt < 4; ++nt) {
            v2f b;
            int bidx = (k + 2 * half) * LDSPB + nt * 16 + m;
            b.x = Xs[bidx];
            b.y = Xs[bidx + LDSPB];
            accRe[nt] = __builtin_amdgcn_wmma_f32_16x16x4_f32(
                false, aRe, false, b, (short)0, accRe[nt], false, false);
            accIm[nt] = __builtin_amdgcn_wmma_f32_16x16x4_f32(
                false, aIm, false, b, (short)0, accIm[nt], false, false);
        }
    }

    size_t outBase = (size_t)bl * kC * kHW;
#pragma unroll
    for (int nt = 0; nt < 4; ++nt) {
        int n = col0 + nt * 16 + m;
#pragma unroll
        for (int r = 0; r < 8; ++r) {
            int row = rowBase + r + 8 * half;
            Dre[outBase + (size_t)row * kHW + n] = accRe[nt][r];
            Dim[outBase + (size_t)row * kHW + n] = accIm[nt][r];
        }
    }
}

// ----------------------------------------------------- FFT2 (32x32, c2c) ----
// mode 0: forward + epilogue (val + bb) * gamma[o,h]
// mode 1: inverse with 1/1024 scaling
__global__ __launch_bounds__(256) void convlru_fft2(
        float* __restrict__ Re, float* __restrict__ Im,
        const float* __restrict__ bbre, const float* __restrict__ bbim,
        const float* __restrict__ gam, int mode) {
    __shared__ float ar[1024], ai[1024], br[1024], bi[1024];
    __shared__ float twr[32], twi[32];

    const int g = blockIdx.x;            // bl*C + o
    const int o = g & (kC - 1);
    const size_t base = (size_t)g * kHW;
    const float sgn = (mode == 0) ? -1.f : 1.f;

    if (threadIdx.x < 32) {
        float s, c;
        __sincosf(sgn * 6.28318530717958647692f * (float)threadIdx.x / 32.f, &s, &c);
        twr[threadIdx.x] = c;
        twi[threadIdx.x] = s;
    }
    for (int i = threadIdx.x; i < 1024; i += 256) {
        ar[i] = Re[base + i];
        ai[i] = Im[base + i];
    }
    __syncthreads();

    // DFT along W (rows): b[r,k] = sum_n a[r,n] * w^{nk}
    for (int i = threadIdx.x; i < 1024; i += 256) {
        int r = i >> 5, k = i & 31;
        const float* rr = ar + r * 32;
        const float* ri = ai + r * 32;
        float sr = 0.f, si = 0.f;
        for (int n = 0; n < 32; ++n) {
            int t = (n * k) & 31;
            float wr = twr[t], wi = twi[t];
            sr += rr[n] * wr - ri[n] * wi;
            si += rr[n] * wi + ri[n] * wr;
        }
        br[i] = sr; bi[i] = si;
    }
    __syncthreads();

    // DFT along H (cols): out[k,c] = sum_n b[n,c] * w^{nk}
    for (int i = threadIdx.x; i < 1024; i += 256) {
        int k = i >> 5, c = i & 31;
        float sr = 0.f, si = 0.f;
        for (int n = 0; n < 32; ++n) {
            int t = (n * k) & 31;
            float wr = twr[t], wi = twi[t];
            float xr = br[n * 32 + c], xi = bi[n * 32 + c];
            sr += xr * wr - xi * wi;
            si += xr * wi + xi * wr;
        }
        if (mode == 0) {
            float gm = gam[o * kH + k];
            sr = (sr + bbre[o]) * gm;
            si = (si + bbim[o]) * gm;
        } else {
            sr *= (1.f / 1024.f);
            si *= (1.f / 1024.f);
        }
        Re[base + i] = sr;
        Im[base + i] = si;
    }
}

// -------------------------------------------------------------- LRU scan ----
__global__ void convlru_scan(float* __restrict__ Dre, float* __restrict__ Dim,
                             const float* __restrict__ lamRe,
                             const float* __restrict__ lamIm,
                             float* __restrict__ lastOut) {
    int idx = blockIdx.x * blockDim.x + threadIdx.x;   // B*C*HW chains
    if (idx >= kB * kC * kHW) return;
    int hw = idx & (kHW - 1);
    int o  = (idx >> 10) & (kC - 1);
    int b  = idx >> 17;
    int hrow = hw >> 5;
    float lr = lamRe[o * kH + hrow], li = lamIm[o * kH + hrow];
    float yr = 0.f, yi = 0.f;
    for (int l = 0; l < kL; ++l) {
        size_t a = ((size_t)(b * kL + l) * kC + o) * kHW + hw;
        if (l + 1 < kL) {                    // hide the 512KB l-stride in L2
            size_t an = a + (size_t)kC * kHW;
            __builtin_prefetch(&Dre[an], 0, 1);
            __builtin_prefetch(&Dim[an], 0, 1);
        }
        float hr = Dre[a], hi = Dim[a];
        float nyr = lr * yr - li * yi + hr;
        float nyi = lr * yi + li * yr + hi;
        yr = nyr; yi = nyi;
        Dre[a] = yr; Dim[a] = yi;
    }
    size_t oidx = (((size_t)b * kC + o) * kHW + hw) * 2;  // complex64 interleaved
    lastOut[oidx]     = yr;
    lastOut[oidx + 1] = yi;
}

// ------------------------------------------------------------- mixC GEMM ----
// E[bl,o,hw] = sum_c Wc_re[o,c]*Sre - Wc_im[o,c]*Sim + bc_re[o]
// Two TDM descriptors stage the Sre/Sim tiles (pad -> stride 33).
#define LDSPC 33
__global__ __launch_bounds__(256) void convlru_mixc(
        const float* __restrict__ Sre, const float* __restrict__ Sim,
        const float* __restrict__ Wre, const float* __restrict__ Wim,
        const float* __restrict__ bcre, float* __restrict__ E) {
    __shared__ float XsR[128 * LDSPC];
    __shared__ float XsI[128 * LDSPC];
    const int bl   = blockIdx.y;
    const int col0 = blockIdx.x * 32;
    const size_t ib = (size_t)bl * kC * kHW;

    if (threadIdx.x < 32) {       // wave 0 issues both DMAs, waits TENSORcnt==0
        unsigned long long gaR =
            (unsigned long long)(const void*)(Sre + ib + col0);
        unsigned long long gaI =
            (unsigned long long)(const void*)(Sim + ib + col0);
        tdm_load_tile_f32((unsigned)(unsigned long long)(const void*)XsR, gaR,
                          kHW, kC, kHW, 32u, 128u, /*pad 32DW*/ 4u);
        tdm_load_tile_f32((unsigned)(unsigned long long)(const void*)XsI, gaI,
                          kHW, kC, kHW, 32u, 128u, /*pad 32DW*/ 4u);
        __builtin_amdgcn_s_wait_tensorcnt(0);
    }
    __syncthreads();

    const int wave = threadIdx.x >> 5;
    const int lane = threadIdx.x & 31;
    const int rowBase = wave * 16;
    const int m = lane & 15;
    const int half = lane >> 4;

    v8f acc[2] = {};

    for (int k = 0; k < 128; k += 4) {
        v2f aRe, aIm;
        int arow = (rowBase + m) * kC + k + 2 * half;
        aRe.x =  Wre[arow]; aRe.y =  Wre[arow + 1];
        aIm.x = -Wim[arow]; aIm.y = -Wim[arow + 1];   // fold the minus sign
#pragma unroll
        for (int nt = 0; nt < 2; ++nt) {
            v2f bR, bI;
            int bidx = (k + 2 * half) * LDSPC + nt * 16 + m;
            bR.x = XsR[bidx]; bR.y = XsR[bidx + LDSPC];
            bI.x = XsI[bidx]; bI.y = XsI[bidx + LDSPC];
            acc[nt] = __builtin_amdgcn_wmma_f32_16x16x4_f32(
                false, aRe, false, bR, (short)0, acc[nt], false, false);
            acc[nt] = __builtin_amdgcn_wmma_f32_16x16x4_f32(
                false, aIm, false, bI, (short)0, acc[nt], false, false);
        }
    }

#pragma unroll
    for (int nt = 0; nt < 2; ++nt) {
        int n = col0 + nt * 16 + m;
#pragma unroll
        for (int r = 0; r < 8; ++r) {
            int row = rowBase + r + 8 * half;
            E[ib + (size_t)row * kHW + n] = acc[nt][r] + bcre[row];
        }
    }
}

// ------------------------------------------------------ LayerNorm + res ----
__global__ __launch_bounds__(1024) void convlru_ln(
        const float* __restrict__ E, const float* __restrict__ x,
        const float* __restrict__ lnw, const float* __restrict__ lnb,
        float* __restrict__ out) {
    __shared__ float red[1024];
    __shared__ float red2[1024];
    const int bl = blockIdx.x;
    const size_t base = (size_t)bl * (kC * kHW);
    float s = 0.f, s2 = 0.f;
    for (int i = threadIdx.x; i < kC * kHW; i += 1024) {
        float v = E[base + i];
        s += v; s2 += v * v;
    }
    red[threadIdx.x] = s;
    red2[threadIdx.x] = s2;
    __syncthreads();
    for (int st = 512; st > 0; st >>= 1) {
        if ((int)threadIdx.x < st) {
            red[threadIdx.x]  += red[threadIdx.x + st];
            red2[threadIdx.x] += red2[threadIdx.x + st];
        }
        __syncthreads();
    }
    const float inv_n = 1.f / (float)(kC * kHW);
    float mean = red[0] * inv_n;
    float var  = red2[0] * inv_n - mean * mean;
    float inv  = rsqrtf(var + 1e-5f);
    for (int i = threadIdx.x; i < kC * kHW; i += 1024) {
        out[base + i] = (E[base + i] - mean) * inv * lnw[i] + lnb[i] + x[base + i];
    }
}

// ----------------------------------------------------------------- launch ----
extern "C" void kernel_launch(void* const* d_in, const int* in_sizes, int n_in,
                              void* d_out, int out_size, void* d_ws, size_t ws_size,
                              hipStream_t stream) {
    const float* x     = (const float*)d_in[0];
    const float* plog  = (const float*)d_in[1];
    const float* Wb_re = (const float*)d_in[2];
    const float* Wb_im = (const float*)d_in[3];
    const float* bb_re = (const float*)d_in[4];
    const float* bb_im = (const float*)d_in[5];
    const float* Wc_re = (const float*)d_in[6];
    const float* Wc_im = (const float*)d_in[7];
    const float* bc_re = (const float*)d_in[8];
    // d_in[9] = bc_im (unused: only real part of output kept)
    const float* ln_w  = (const float*)d_in[10];
    const float* ln_b  = (const float*)d_in[11];

    float* ws    = (float*)d_ws;
    float* Dre   = ws;                 // kN1 floats
    float* Dim   = ws + (size_t)kN1;   // kN1 floats
    float* E     = ws + 2 * (size_t)kN1;
    float* lamRe = ws + 3 * (size_t)kN1;
    float* lamIm = lamRe + kCH;
    float* gam   = lamIm + kCH;

    float* out1 = (float*)d_out;          // (B,L,C,H,W) f32
    float* out2 = out1 + (size_t)kN1;     // (B,1,C,H,W) complex64 interleaved

    convlru_params<<<(kCH + 255) / 256, 256, 0, stream>>>(plog, lamRe, lamIm, gam);
    convlru_mixb<<<dim3(kHW / 64, kB * kL), 256, 0, stream>>>(x, Wb_re, Wb_im, Dre, Dim);
    convlru_fft2<<<kB * kL * kC, 256, 0, stream>>>(Dre, Dim, bb_re, bb_im, gam, 0);
    convlru_scan<<<(kB * kC * kHW) / 256, 256, 0, stream>>>(Dre, Dim, lamRe, lamIm, out2);
    convlru_fft2<<<kB * kL * kC, 256, 0, stream>>>(Dre, Dim, bb_re, bb_im, gam, 1);
    convlru_mixc<<<dim3(kHW / 32, kB * kL), 256, 0, stream>>>(Dre, Dim, Wc_re, Wc_im, bc_re, E);
    convlru_ln<<<kB * kL, 1024, 0, stream>>>(E, x, ln_w, ln_b, out1);
}